// KGAggregator_25280177504545
// MI455X (gfx1250) — compile-verified
//
#include <hip/hip_runtime.h>

#define N_ENT   100000
#define DIM     128
#define N_EDGE  1600000
#define SLOPE   0.01f
#define LSTRIDE 132   // 128 + 4 dword pad: bank = (4*row + col) % 64 -> conflict-free

typedef __attribute__((ext_vector_type(2))) float v2f;
typedef __attribute__((ext_vector_type(4))) float v4f;
typedef __attribute__((ext_vector_type(8))) float v8f;
typedef __attribute__((ext_vector_type(4))) unsigned int u32x4;
typedef __attribute__((ext_vector_type(8))) int i32x8;
typedef __attribute__((ext_vector_type(4))) int i32x4;

// LDS layout inside one __shared__ block (byte offsets)
#define WS_OFF   0
#define WN_OFF   (DIM * LSTRIDE * 4)              // 67584
#define XT_OFF   (2 * DIM * LSTRIDE * 4)          // 135168
#define NT_OFF   (XT_OFF + 16 * LSTRIDE * 4)      // 143616
#define SMEM_FLOATS (2 * DIM * LSTRIDE + 2 * 16 * LSTRIDE)  // 38016 -> 152064 B

// ---------------------------------------------------------------- zero scratch
__global__ void kg_zero(float* __restrict__ ws, int n) {
    int i = (blockIdx.x * blockDim.x + threadIdx.x) * 4;
    const int stride = gridDim.x * blockDim.x * 4;
    v4f z = {0.f, 0.f, 0.f, 0.f};
    for (; i < n; i += stride) *(v4f*)(ws + i) = z;
}

// ------------------------------------------------- COO SpMM scatter (1 wave/edge)
// neigh[r, :] += v * x[c, :]   (128 dims = 32 lanes x float4, 4 f32 atomics/lane)
__global__ void kg_scatter(const float* __restrict__ x,
                           const int*   __restrict__ rows,
                           const int*   __restrict__ cols,
                           const float* __restrict__ vals,
                           float*       __restrict__ acc) {
    const int lane   = threadIdx.x & 31;
    const int wave   = blockIdx.x * (blockDim.x >> 5) + (threadIdx.x >> 5);
    const int nwaves = gridDim.x * (blockDim.x >> 5);
    for (int e = wave; e < N_EDGE; e += nwaves) {
        const int   r = rows[e];
        const int   c = cols[e];
        const float v = vals[e];
        v4f xa = *(const v4f*)(x + (size_t)c * DIM + lane * 4);
        float* dst = acc + (size_t)r * DIM + lane * 4;
        atomicAdd(dst + 0, v * xa.x);
        atomicAdd(dst + 1, v * xa.y);
        atomicAdd(dst + 2, v * xa.z);
        atomicAdd(dst + 3, v * xa.w);
    }
}

#if __has_builtin(__builtin_amdgcn_tensor_load_to_lds)
#define HAVE_TDM 1
// DMA a [rows x 128] f32 tile (row-major, contiguous rows) into LDS at lds_addr,
// inserting 4 pad dwords after every 128 dwords -> LSTRIDE=132 rows in LDS.
// D# per CDNA5 ISA ch.8: group0 = {count|flags, lds_addr, gaddr_lo, gaddr_hi|type},
// group1 = {pad/data_size, dim0, dim1, tile0, tile1|tile2, stride0_lo, stride0_hi|stride1, 0}.
__device__ __forceinline__ void tdm_load_tile(const float* gsrc, unsigned lds_addr, int rows) {
    unsigned long long ga = (unsigned long long)(uintptr_t)gsrc;
    u32x4 g0 = {1u,                                  // count=1, is_restore=0, gather=0
                lds_addr,                            // LDS byte address
                (unsigned)ga,                        // global addr [31:0]
                (unsigned)(ga >> 32) | (2u << 30)};  // global addr [56:32], type=2
    i32x8 g1 = {(int)0x07920000,                     // data_size=4B, pad_en, intvl=128dw, amt=4dw
                (int)(128u << 16),                   // tensor_dim0 = 128
                (int)((unsigned)rows << 16),         // tensor_dim1 = rows
                (int)(128u << 16),                   // tile_dim0 = 128
                rows,                                // tile_dim1 = rows (tile_dim2 = 0)
                128,                                 // tensor_dim0_stride = 128
                0, 0};
    i32x4 z4 = {0, 0, 0, 0};
    i32x8 z8 = {0, 0, 0, 0, 0, 0, 0, 0};
    // amdgpu-toolchain (clang-23) 6-arg form: (g0, g1, g2, g3, g4, cpol)
    __builtin_amdgcn_tensor_load_to_lds(g0, g1, z4, z4, z8, 0);
}
#else
#define HAVE_TDM 0
#endif

// -------------------------------------- fused dual-GEMM + bias + LeakyReLU (WMMA f32)
// out = leaky( x @ Ws^T + neigh @ Wn^T + (bs + bn) )
// Block = 256 threads = 8 wave32; wave w owns output columns [16w, 16w+16).
__global__ void __launch_bounds__(256) kg_gemm(const float* __restrict__ x,
                                               const float* __restrict__ neigh,
                                               const float* __restrict__ Ws,
                                               const float* __restrict__ Wn,
                                               const float* __restrict__ bs,
                                               const float* __restrict__ bn,
                                               float* __restrict__ out) {
    __shared__ float smem[SMEM_FLOATS];
    float* lds_ws = smem;
    float* lds_wn = smem + DIM * LSTRIDE;
    float* lds_x  = smem + 2 * DIM * LSTRIDE;
    float* lds_n  = lds_x + 16 * LSTRIDE;

    const int tid  = threadIdx.x;
    const int lane = tid & 31;
    const int wave = tid >> 5;

#if HAVE_TDM
    // Generic-pointer low 32 bits == LDS byte offset (ISA aperture rule).
    const unsigned lds_base = (unsigned)(uintptr_t)(void*)smem;
    if (wave == 0) {   // TDM ignores EXEC; one wave issues the whole-matrix DMAs
        tdm_load_tile(Ws, lds_base + WS_OFF, DIM);
        tdm_load_tile(Wn, lds_base + WN_OFF, DIM);
    }
#else
    for (int idx = tid * 4; idx < DIM * DIM; idx += 256 * 4) {
        const int row = idx >> 7, col = idx & 127;
        *(v4f*)&lds_ws[row * LSTRIDE + col] = *(const v4f*)(Ws + idx);
        *(v4f*)&lds_wn[row * LSTRIDE + col] = *(const v4f*)(Wn + idx);
    }
#endif

    const int jl    = lane & 15;            // N index within tile / A-matrix row
    const int j     = wave * 16 + jl;       // global output column
    const int khalf = (lane >> 4) * 2;      // K sub-offset: lanes 0-15 -> {0,1}, 16-31 -> {2,3}
    const float bias = bs[j] + bn[j];

    const int ntiles = N_ENT / 16;          // 6250
    for (int tile = blockIdx.x; tile < ntiles; tile += gridDim.x) {
        const size_t row0 = (size_t)tile * 16;

        __syncthreads();   // previous iteration's readers are done with LDS tiles
#if HAVE_TDM
        if (wave == 0) {
            tdm_load_tile(x     + row0 * DIM, lds_base + XT_OFF, 16);
            tdm_load_tile(neigh + row0 * DIM, lds_base + NT_OFF, 16);
            __builtin_amdgcn_s_wait_tensorcnt(0);   // also covers the W DMAs (in-order)
        }
#else
        for (int idx = tid * 4; idx < 16 * DIM; idx += 256 * 4) {
            const int row = idx >> 7, col = idx & 127;
            *(v4f*)&lds_x[row * LSTRIDE + col] = *(const v4f*)(x     + row0 * DIM + idx);
            *(v4f*)&lds_n[row * LSTRIDE + col] = *(const v4f*)(neigh + row0 * DIM + idx);
        }
#endif
        __syncthreads();

        v8f c = {0.f, 0.f, 0.f, 0.f, 0.f, 0.f, 0.f, 0.f};
        // ISA f32 A 16x4 layout: lane l<16 -> (K,K+1); l>=16 -> (K+2,K+3); M = l%16.
        // B 4x16 mirrored: VGPR pair holds (B[K+khalf][j], B[K+khalf+1][j]) = W[j][K+khalf..].
#pragma unroll 4
        for (int k = 0; k < DIM; k += 4) {
            v2f a_s = *(const v2f*)&lds_x [jl * LSTRIDE + k + khalf];
            v2f b_s = *(const v2f*)&lds_ws[j  * LSTRIDE + k + khalf];
            c = __builtin_amdgcn_wmma_f32_16x16x4_f32(false, a_s, false, b_s,
                                                      (short)0, c, false, false);
            v2f a_n = *(const v2f*)&lds_n [jl * LSTRIDE + k + khalf];
            v2f b_n = *(const v2f*)&lds_wn[j  * LSTRIDE + k + khalf];
            c = __builtin_amdgcn_wmma_f32_16x16x4_f32(false, a_n, false, b_n,
                                                      (short)0, c, false, false);
        }

        // C/D 16x16 layout: VGPR v -> M = v + 8*(lane>=16), N = lane%16.
        const int m_off = (lane >> 4) * 8;
        float* orow = out + row0 * DIM + j;
#pragma unroll
        for (int v = 0; v < 8; ++v) {
            float o = c[v] + bias;
            o = (o > 0.f) ? o : SLOPE * o;
            orow[(size_t)(v + m_off) * DIM] = o;
        }
    }
}

// ---------------------------------------------------------------------- launch
extern "C" void kernel_launch(void* const* d_in, const int* in_sizes, int n_in,
                              void* d_out, int out_size, void* d_ws, size_t ws_size,
                              hipStream_t stream) {
    const float* x    = (const float*)d_in[0];
    const int*   rows = (const int*)  d_in[1];
    const int*   cols = (const int*)  d_in[2];
    const float* vals = (const float*)d_in[3];
    const float* Ws   = (const float*)d_in[4];
    const float* bs   = (const float*)d_in[5];
    const float* Wn   = (const float*)d_in[6];
    const float* bn   = (const float*)d_in[7];
    float* out = (float*)d_out;
    float* acc = (float*)d_ws;   // N_ENT * DIM floats = 51.2 MB scratch accumulator

    kg_zero   <<<2048, 256, 0, stream>>>(acc, N_ENT * DIM);
    kg_scatter<<<4096, 256, 0, stream>>>(x, rows, cols, vals, acc);
    kg_gemm   <<< 512, 256, 0, stream>>>(x, acc, Ws, Wn, bs, bn, out);
}